// SimpleGCN_53197464928387
// MI455X (gfx1250) — compile-verified
//
#include <hip/hip_runtime.h>
#include <hip/hip_bf16.h>

typedef __attribute__((ext_vector_type(2))) float v2f;
typedef __attribute__((ext_vector_type(8))) float v8f;

// ---------------------------------------------------------------------------
// GEMM: Out[M,N] = A[M,K] @ W[K,N]   (fp32, V_WMMA_F32_16X16X4_F32)
// One wave32 -> 16x64 output tile (4 accumulators). blockDim = 64 (2 waves),
// so one block covers a 16x128 tile. M multiple of 16, N multiple of 128,
// K multiple of 4 (all guaranteed by the fixed problem dims).
//
// A-operand layout (ISA 7.12.2, 32-bit A 16x4): lane&15 = M row,
// lane>>4 selects K-pair {0,1} / {2,3}; pair is contiguous -> b64 load.
// B-operand mirrors with lane&15 = N col. C/D: VGPR v holds M=v (lanes 0-15)
// and M=8+v (lanes 16-31), N = lane&15.
// ---------------------------------------------------------------------------
__global__ __launch_bounds__(64) void gcn_gemm_wmma_f32(
    const float* __restrict__ A, int lda,
    const float* __restrict__ W, int ldw,
    float* __restrict__ Out, int ldo,
    int K)
{
  const int lane = threadIdx.x & 31;
  const int wave = threadIdx.x >> 5;
  const int l15  = lane & 15;
  const int lh   = lane >> 4;            // 0 -> K {0,1}, 1 -> K {2,3}
  const int m0   = blockIdx.x * 16;
  const int n0   = blockIdx.y * 128 + wave * 64;

  v8f acc0 = {}, acc1 = {}, acc2 = {}, acc3 = {};
  const float* arow = A + (size_t)(m0 + l15) * (size_t)lda + 2 * lh;

  for (int k = 0; k < K; k += 4) {
    v2f a = *(const v2f*)(arow + k);                       // 8B aligned
    const float* wp = W + (size_t)(k + 2 * lh) * (size_t)ldw + n0 + l15;
    v2f b0, b1, b2, b3;
    b0.x = wp[0];   b0.y = wp[ldw];
    b1.x = wp[16];  b1.y = wp[16 + ldw];
    b2.x = wp[32];  b2.y = wp[32 + ldw];
    b3.x = wp[48];  b3.y = wp[48 + ldw];
    acc0 = __builtin_amdgcn_wmma_f32_16x16x4_f32(false, a, false, b0, (short)0, acc0, false, false);
    acc1 = __builtin_amdgcn_wmma_f32_16x16x4_f32(false, a, false, b1, (short)0, acc1, false, false);
    acc2 = __builtin_amdgcn_wmma_f32_16x16x4_f32(false, a, false, b2, (short)0, acc2, false, false);
    acc3 = __builtin_amdgcn_wmma_f32_16x16x4_f32(false, a, false, b3, (short)0, acc3, false, false);
  }

  const int rbase = m0 + lh * 8;
#pragma unroll
  for (int v = 0; v < 8; ++v) {
    float* orow = Out + (size_t)(rbase + v) * (size_t)ldo + n0 + l15;
    orow[0]  = acc0[v];
    orow[16] = acc1[v];
    orow[32] = acc2[v];
    orow[48] = acc3[v];
  }
}

// ---------------------------------------------------------------------------
// Degree / normalization helpers
// ---------------------------------------------------------------------------
__global__ void gcn_deg_init(float* __restrict__ deg, int n) {
  int i = blockIdx.x * blockDim.x + threadIdx.x;
  if (i < n) deg[i] = 1.0f;                 // self-loop contributes 1
}

__global__ void gcn_deg_accum(const int* __restrict__ dst, int E,
                              float* __restrict__ deg) {
  int e = blockIdx.x * blockDim.x + threadIdx.x;
  if (e < E) atomicAdd(&deg[dst[e]], 1.0f);
}

__global__ void gcn_deg_rsqrt(float* __restrict__ deg, int n) {
  int i = blockIdx.x * blockDim.x + threadIdx.x;
  if (i < n) deg[i] = __frsqrt_rn(deg[i]);  // deg >= 1 always
}

// ---------------------------------------------------------------------------
// Initialize an output column-slice to the bias (aggregation base value).
// ---------------------------------------------------------------------------
__global__ void gcn_bias_init(float* __restrict__ out, int ldo, int coloff,
                              const float* __restrict__ b, int C, long total) {
  long t = (long)blockIdx.x * blockDim.x + threadIdx.x;
  if (t >= total) return;
  long i = t / C;
  int  c = (int)(t - i * C);
  out[i * (long)ldo + coloff + c] = b[c];
}

// ---------------------------------------------------------------------------
// Edge scatter: out[dst, coloff+c] += h[src, c] * dinv[src]*dinv[dst]
// Blocks [0,E) are real edges; blocks [E, E+Nn) are self-loops (src=dst=i).
// blockDim = C (128 or 256) -> coalesced gather of h row, per-wave coalesced
// atomics that resolve in L2 (working set << 192 MB).
// ---------------------------------------------------------------------------
__global__ void gcn_scatter(const int* __restrict__ src,
                            const int* __restrict__ dst, int E,
                            const float* __restrict__ dinv,
                            const float* __restrict__ h, int ldh,
                            float* __restrict__ out, int ldo, int coloff) {
  int e = blockIdx.x;
  int c = threadIdx.x;
  int s, d;
  if (e < E) { s = src[e]; d = dst[e]; }
  else       { s = d = e - E; }
  float norm = dinv[s] * dinv[d];
  float val  = h[(long)s * ldh + c] * norm;
  atomicAdd(&out[(long)d * ldo + coloff + c], val);
}

// ---------------------------------------------------------------------------
// ReLU over an output column-slice.
// ---------------------------------------------------------------------------
__global__ void gcn_relu(float* __restrict__ out, int ldo, int coloff,
                         int C, long total) {
  long t = (long)blockIdx.x * blockDim.x + threadIdx.x;
  if (t >= total) return;
  long i = t / C;
  int  c = (int)(t - i * C);
  float* p = &out[i * (long)ldo + coloff + c];
  *p = fmaxf(*p, 0.0f);
}

// ---------------------------------------------------------------------------
// Orchestration
// ---------------------------------------------------------------------------
extern "C" void kernel_launch(void* const* d_in, const int* in_sizes, int n_in,
                              void* d_out, int out_size, void* d_ws, size_t ws_size,
                              hipStream_t stream) {
  const float* x  = (const float*)d_in[0];
  const int*   ei = (const int*)d_in[1];
  const float* W1 = (const float*)d_in[2];
  const float* b1 = (const float*)d_in[3];
  const float* W2 = (const float*)d_in[4];
  const float* b2 = (const float*)d_in[5];
  const float* W3 = (const float*)d_in[6];
  const float* b3 = (const float*)d_in[7];
  float* out = (float*)d_out;

  const int IN_C = 256, H2 = 256, H1 = 128, OUT_C = 128;
  const int LDO  = H2 + H1 + OUT_C;          // 512: concat(h1,h2,h3)
  const int Nn   = in_sizes[0] / IN_C;       // 50000 (multiple of 16)
  const int E    = in_sizes[1] / 2;          // 800000
  const int* src = ei;
  const int* dst = ei + E;

  // workspace: dinv [Nn] floats, then raw-h buffer [Nn x 256] floats
  float* dinv = (float*)d_ws;
  float* hbuf = (float*)((char*)d_ws + (((size_t)Nn * sizeof(float) + 255) & ~(size_t)255));

  const int TB = 256;
  const int mtiles = Nn / 16;

  // ---- symmetric normalization: dinv = rsqrt(1 + in-degree) ----
  gcn_deg_init <<<(Nn + TB - 1) / TB, TB, 0, stream>>>(dinv, Nn);
  gcn_deg_accum<<<(E  + TB - 1) / TB, TB, 0, stream>>>(dst, E, dinv);
  gcn_deg_rsqrt<<<(Nn + TB - 1) / TB, TB, 0, stream>>>(dinv, Nn);

  // ---- layer 1: h = x@W1 ; h1 = relu(agg(h) + b1) -> out[:, 0:256] ----
  {
    long tot = (long)Nn * H2;
    gcn_gemm_wmma_f32<<<dim3(mtiles, H2 / 128), 64, 0, stream>>>(x, IN_C, W1, H2, hbuf, H2, IN_C);
    gcn_bias_init<<<(unsigned)((tot + TB - 1) / TB), TB, 0, stream>>>(out, LDO, 0, b1, H2, tot);
    gcn_scatter  <<<E + Nn, H2, 0, stream>>>(src, dst, E, dinv, hbuf, H2, out, LDO, 0);
    gcn_relu     <<<(unsigned)((tot + TB - 1) / TB), TB, 0, stream>>>(out, LDO, 0, H2, tot);
  }

  // ---- layer 2: h = h1@W2 ; h2 = relu(agg(h) + b2) -> out[:, 256:384] ----
  {
    long tot = (long)Nn * H1;
    gcn_gemm_wmma_f32<<<dim3(mtiles, H1 / 128), 64, 0, stream>>>(out, LDO, W2, H1, hbuf, H1, H2);
    gcn_bias_init<<<(unsigned)((tot + TB - 1) / TB), TB, 0, stream>>>(out, LDO, H2, b2, H1, tot);
    gcn_scatter  <<<E + Nn, H1, 0, stream>>>(src, dst, E, dinv, hbuf, H1, out, LDO, H2);
    gcn_relu     <<<(unsigned)((tot + TB - 1) / TB), TB, 0, stream>>>(out, LDO, H2, H1, tot);
  }

  // ---- layer 3: h = h2@W3 ; h3 = relu(agg(h) + b3) -> out[:, 384:512] ----
  {
    long tot = (long)Nn * OUT_C;
    gcn_gemm_wmma_f32<<<dim3(mtiles, OUT_C / 128), 64, 0, stream>>>(out + H2, LDO, W3, OUT_C, hbuf, OUT_C, H1);
    gcn_bias_init<<<(unsigned)((tot + TB - 1) / TB), TB, 0, stream>>>(out, LDO, H2 + H1, b3, OUT_C, tot);
    gcn_scatter  <<<E + Nn, OUT_C, 0, stream>>>(src, dst, E, dinv, hbuf, OUT_C, out, LDO, H2 + H1);
    gcn_relu     <<<(unsigned)((tot + TB - 1) / TB), TB, 0, stream>>>(out, LDO, H2 + H1, OUT_C, tot);
  }
}